// KatiesDecoder_51470888075939
// MI455X (gfx1250) — compile-verified
//
#include <hip/hip_runtime.h>

// KatiesDecoder gather: out[b, v, k*128 + d] = z_prime[b, index[v,k], d]
// Pure bandwidth-bound gather-copy. One wave32 per (v,k) group; each lane
// moves one float4 (16B) per batch; 4 batches per wave amortize the index
// fetch. NT stores keep the 84MB z_prime source resident in the 192MB L2.

typedef float v4f __attribute__((ext_vector_type(4)));

static constexpr int D_LAT      = 128;         // latent dim
static constexpr int NU         = 3;           // rows gathered per vertex
static constexpr int BATCH      = 4;           // B
static constexpr int V4_PER_ROW = D_LAT / 4;   // 32 == wave32 lanes

__global__ __launch_bounds__(256) void KatiesDecoder_gather_kernel(
    const float* __restrict__ z_prime,   // [B, n_dual, 128]
    const int*   __restrict__ index,     // [n_vertex * 3] (int32, flat groups)
    float*       __restrict__ out,       // [B, n_vertex, 3*128]
    long long    ngroups,                // n_vertex * 3
    int          n_dual)
{
    const int       lane  = threadIdx.x & 31;
    const long long group = (long long)blockIdx.x * (blockDim.x >> 5)
                          + (threadIdx.x >> 5);
    if (group >= ngroups) return;

    // Prefetch the index stream well ahead of the wavefront (global_prefetch_b8).
    const long long AHEAD = 16384;
    if (group + AHEAD < ngroups) {
        __builtin_prefetch(&index[group + AHEAD], 0, 3);
    }

    // Gather index is uniform across the wave -> scalarize it so the row
    // addresses become SGPR-base + lane-offset (coalesced b128 accesses).
    const int j = __builtin_amdgcn_readfirstlane(index[group]);

    const v4f* __restrict__ src = (const v4f*)z_prime;
    v4f*       __restrict__ dst = (v4f*)out;

    const long long sstride = (long long)n_dual * V4_PER_ROW;  // per-batch src stride (v4f units)
    const long long dstride = ngroups * V4_PER_ROW;            // per-batch dst stride (v4f units)
    const long long srow    = (long long)j * V4_PER_ROW + lane;
    const long long drow    = group * V4_PER_ROW + lane;

    // Issue all 4 loads (default/RT temporal hint: keep z_prime hot in L2),
    // then 4 non-temporal stores (1GB write-once stream must not evict L2).
    v4f r0 = src[srow + 0 * sstride];
    v4f r1 = src[srow + 1 * sstride];
    v4f r2 = src[srow + 2 * sstride];
    v4f r3 = src[srow + 3 * sstride];

    __builtin_nontemporal_store(r0, &dst[drow + 0 * dstride]);
    __builtin_nontemporal_store(r1, &dst[drow + 1 * dstride]);
    __builtin_nontemporal_store(r2, &dst[drow + 2 * dstride]);
    __builtin_nontemporal_store(r3, &dst[drow + 3 * dstride]);
}

extern "C" void kernel_launch(void* const* d_in, const int* in_sizes, int n_in,
                              void* d_out, int out_size, void* d_ws, size_t ws_size,
                              hipStream_t stream) {
    // setup_inputs order: z_prime (f32), x_ancil (f32, unused), index (int32)
    const float* z_prime = (const float*)d_in[0];
    const int*   index   = (const int*)d_in[2];
    float*       out     = (float*)d_out;

    const long long z_elems = (long long)in_sizes[0];          // B * n_dual * 128
    const long long ngroups = (long long)in_sizes[2];          // n_vertex * 3
    const int       n_dual  = (int)(z_elems / ((long long)BATCH * D_LAT));

    const int  threads         = 256;                 // 8 waves per block
    const int  waves_per_block = threads / 32;
    const long long nblocks = (ngroups + waves_per_block - 1) / waves_per_block;

    KatiesDecoder_gather_kernel<<<dim3((unsigned)nblocks), dim3(threads), 0, stream>>>(
        z_prime, index, out, ngroups, n_dual);
}